// Net1_16587163698027
// MI455X (gfx1250) — compile-verified
//
#include <hip/hip_runtime.h>
#include <hip/hip_bf16.h>

#define N_NODES 50000
#define N_EDGES 800000
#define N_GRAPHS 256
#define D_IN 14
#define D_HID 256

typedef __attribute__((ext_vector_type(2))) float v2f;
typedef __attribute__((ext_vector_type(8))) float v8f;

__device__ __forceinline__ float selu_f(float x) {
    const float scale = 1.0507009873554805f;
    const float alpha = 1.6732632423543772f;
    return scale * (x > 0.0f ? x : alpha * (expf(x) - 1.0f));
}

// ---------------- utility ----------------
__global__ void k_zero(float* p, int n) {
    int i = blockIdx.x * blockDim.x + threadIdx.x;
    if (i < n) p[i] = 0.0f;
}

__global__ void k_deg_count(const int* __restrict__ dst, float* __restrict__ deg) {
    int e = blockIdx.x * blockDim.x + threadIdx.x;
    if (e < N_EDGES) atomicAdd(&deg[dst[e]], 1.0f);
}

__global__ void k_dinv(const float* __restrict__ deg, float* __restrict__ dinv) {
    int n = blockIdx.x * blockDim.x + threadIdx.x;
    if (n < N_NODES) dinv[n] = rsqrtf(deg[n] + 1.0f);  // +1 self loop; always > 0
}

// ---------------- GEMM1: [N,14] @ [14,256] ----------------
__global__ __launch_bounds__(256) void k_gemm14(const float* __restrict__ x,
                                                const float* __restrict__ W1,
                                                float* __restrict__ H) {
    __shared__ float xs[D_IN];
    int n = blockIdx.x;
    int f = threadIdx.x;
    if (f < D_IN) xs[f] = x[n * D_IN + f];
    __syncthreads();
    float s = 0.0f;
#pragma unroll
    for (int k = 0; k < D_IN; ++k) s += xs[k] * W1[k * D_HID + f];
    H[n * D_HID + f] = s;
}

// ---------------- self-loop init: agg = h * dinv^2 ----------------
__global__ void k_self(const float* __restrict__ H, const float* __restrict__ dinv,
                       float* __restrict__ AGG) {
    int i = blockIdx.x * blockDim.x + threadIdx.x;
    if (i < N_NODES * D_HID) {
        float dv = dinv[i >> 8];
        AGG[i] = H[i] * dv * dv;
    }
}

// ---------------- edge scatter: agg[dst] += h[src] * dinv[src]*dinv[dst] ----------------
__global__ __launch_bounds__(256) void k_scatter(const int* __restrict__ src,
                                                 const int* __restrict__ dst,
                                                 const float* __restrict__ dinv,
                                                 const float* __restrict__ H,
                                                 float* __restrict__ AGG) {
    long long gid = (long long)blockIdx.x * blockDim.x + threadIdx.x;  // E * 64 threads
    int e = (int)(gid >> 6);
    int fg = (int)(gid & 63);
    if (e >= N_EDGES) return;
    int s = src[e], d = dst[e];
    float norm = dinv[s] * dinv[d];
    int f = fg * 4;
    const float4 h4 = *reinterpret_cast<const float4*>(H + (long long)s * D_HID + f);
    float* out = AGG + (long long)d * D_HID + f;
    atomicAdd(out + 0, h4.x * norm);
    atomicAdd(out + 1, h4.y * norm);
    atomicAdd(out + 2, h4.z * norm);
    atomicAdd(out + 3, h4.w * norm);
}

// ---------------- bias + SELU ----------------
__global__ void k_bias_selu(const float* __restrict__ AGG, const float* __restrict__ b,
                            float* __restrict__ X) {
    int i = blockIdx.x * blockDim.x + threadIdx.x;
    if (i < N_NODES * D_HID) X[i] = selu_f(AGG[i] + b[i & (D_HID - 1)]);
}

// ---------------- GEMM2: [N,256] @ [256,256] via V_WMMA_F32_16X16X4_F32 ----------------
// Block = 256 threads (8 waves) handles a 64-row M stripe, half of N (128 cols).
// Wave w owns one 16-col N tile and 4 M sub-tiles -> each B fragment feeds 4 WMMAs.
// Ragged tail handled by shifting the last stripe (duplicate rows get identical values),
// so every load/store is unguarded straight-line code.
__global__ __launch_bounds__(256) void k_gemm_wmma(const float* __restrict__ A,
                                                   const float* __restrict__ B,
                                                   float* __restrict__ C) {
    __shared__ float As[64][D_HID + 4];  // 64x260 floats, rows 16B-aligned
    const int tid = threadIdx.x;
    const int m0 = min((int)blockIdx.x * 64, N_NODES - 64);  // shifted last tile

    // cooperative coalesced float4 load of the 64x256 A tile
#pragma unroll
    for (int i = 0; i < 16; ++i) {
        int f4 = i * 256 + tid;     // 0..4095 float4 slots
        int row = f4 >> 6;          // 0..63
        int col = (f4 & 63) * 4;    // 0..252
        *reinterpret_cast<float4*>(&As[row][col]) =
            *reinterpret_cast<const float4*>(A + (long long)(m0 + row) * D_HID + col);
    }
    __syncthreads();

    const int wave = tid >> 5;
    const int lane = tid & 31;
    const int half = lane >> 4;   // 0: lanes 0-15, 1: lanes 16-31
    const int ml = lane & 15;     // A row within sub-tile / B column within tile
    const int kb = half * 2;      // K sub-offset for this lane half
    const int ncol = blockIdx.y * 128 + wave * 16 + ml;  // global B/C column

    v8f acc[4] = {};
    for (int k0 = 0; k0 < D_HID; k0 += 4) {
        int r = k0 + kb;
        v2f b;
        b[0] = B[(long long)r * D_HID + ncol];
        b[1] = B[(long long)(r + 1) * D_HID + ncol];
#pragma unroll
        for (int mt = 0; mt < 4; ++mt) {
            v2f a = *reinterpret_cast<const v2f*>(&As[mt * 16 + ml][r]);
            acc[mt] = __builtin_amdgcn_wmma_f32_16x16x4_f32(false, a, false, b, (short)0,
                                                            acc[mt], false, false);
        }
    }

#pragma unroll
    for (int mt = 0; mt < 4; ++mt) {
        const int rowbase = m0 + mt * 16 + half * 8;
#pragma unroll
        for (int v = 0; v < 8; ++v)
            C[(long long)(rowbase + v) * D_HID + ncol] = acc[mt][v];
    }
}

// ---------------- global mean pool (accumulate) ----------------
__global__ void k_pool(const float* __restrict__ X, const int* __restrict__ batch,
                       float* __restrict__ gsum, float* __restrict__ gcnt) {
    int i = blockIdx.x * blockDim.x + threadIdx.x;
    if (i < N_NODES * D_HID) {
        int n = i >> 8;
        int f = i & (D_HID - 1);
        int g = batch[n];
        atomicAdd(&gsum[g * D_HID + f], X[i]);
        if (f == 0) atomicAdd(&gcnt[g], 1.0f);
    }
}

// ---------------- MLP head + log_softmax ----------------
__global__ __launch_bounds__(128) void k_head(const float* __restrict__ gsum,
                                              const float* __restrict__ gcnt,
                                              const float* __restrict__ fc1_w,
                                              const float* __restrict__ fc1_b,
                                              const float* __restrict__ fc2_w,
                                              const float* __restrict__ fc2_b,
                                              float* __restrict__ out) {
    __shared__ float sp[D_HID];
    __shared__ float zs[128];
    __shared__ float lg[2];
    int g = blockIdx.x;
    int t = threadIdx.x;
    float cnt = fmaxf(gcnt[g], 1.0f);
    for (int i = t; i < D_HID; i += 128) sp[i] = selu_f(gsum[g * D_HID + i] / cnt);
    __syncthreads();
    float s = fc1_b[t];
    for (int k = 0; k < D_HID; ++k) s += sp[k] * fc1_w[k * 128 + t];
    zs[t] = selu_f(s);
    __syncthreads();
    if (t < 2) {
        float l = fc2_b[t];
        for (int k = 0; k < 128; ++k) l += zs[k] * fc2_w[k * 2 + t];
        lg[t] = l;
    }
    __syncthreads();
    if (t < 2) {
        float m = fmaxf(lg[0], lg[1]);
        float lse = m + logf(expf(lg[0] - m) + expf(lg[1] - m));
        out[g * 2 + t] = lg[t] - lse;
    }
}

extern "C" void kernel_launch(void* const* d_in, const int* in_sizes, int n_in,
                              void* d_out, int out_size, void* d_ws, size_t ws_size,
                              hipStream_t stream) {
    const float* x     = (const float*)d_in[0];
    const int*   eidx  = (const int*)d_in[1];
    const int*   batch = (const int*)d_in[2];
    const float* W1    = (const float*)d_in[3];
    const float* b1    = (const float*)d_in[4];
    const float* W2    = (const float*)d_in[5];
    const float* b2    = (const float*)d_in[6];
    const float* fc1_w = (const float*)d_in[7];
    const float* fc1_b = (const float*)d_in[8];
    const float* fc2_w = (const float*)d_in[9];
    const float* fc2_b = (const float*)d_in[10];
    float* out = (float*)d_out;

    const int* src = eidx;
    const int* dst = eidx + N_EDGES;

    // workspace layout (floats)
    float* ws   = (float*)d_ws;
    float* deg  = ws;                       // 50000
    float* dinv = deg + N_NODES;            // 50000
    float* gcnt = dinv + N_NODES;           // 256
    float* gsum = gcnt + N_GRAPHS;          // 256*256
    float* bufA = gsum + N_GRAPHS * D_HID;  // 50000*256 (16B aligned offset)
    float* bufB = bufA + (long long)N_NODES * D_HID;

    const int NF = N_NODES * D_HID;                 // 12.8M
    const int BLK = 256;
    const int g_nf = (NF + BLK - 1) / BLK;          // 50000
    const int g_nodes = (N_NODES + BLK - 1) / BLK;
    const int g_edges = (N_EDGES + BLK - 1) / BLK;
    const int g_scatter = (N_EDGES * 64) / BLK;     // 200000
    const dim3 g_gemm((N_NODES + 63) / 64, 2);      // 782 x 2

    // degrees & normalization
    k_zero<<<g_nodes, BLK, 0, stream>>>(deg, N_NODES);
    k_deg_count<<<g_edges, BLK, 0, stream>>>(dst, deg);
    k_dinv<<<g_nodes, BLK, 0, stream>>>(deg, dinv);

    // ---- layer 1 ----
    k_gemm14<<<N_NODES, BLK, 0, stream>>>(x, W1, bufA);          // H1 -> A
    k_self<<<g_nf, BLK, 0, stream>>>(bufA, dinv, bufB);          // agg1 -> B
    k_scatter<<<g_scatter, BLK, 0, stream>>>(src, dst, dinv, bufA, bufB);
    k_bias_selu<<<g_nf, BLK, 0, stream>>>(bufB, b1, bufA);       // X1 -> A

    // ---- layer 2 ----
    k_gemm_wmma<<<g_gemm, BLK, 0, stream>>>(bufA, W2, bufB);     // H2 -> B
    k_self<<<g_nf, BLK, 0, stream>>>(bufB, dinv, bufA);          // agg2 -> A
    k_scatter<<<g_scatter, BLK, 0, stream>>>(src, dst, dinv, bufB, bufA);
    k_bias_selu<<<g_nf, BLK, 0, stream>>>(bufA, b2, bufB);       // X2 -> B

    // ---- pool + head ----
    k_zero<<<(N_GRAPHS * D_HID + N_GRAPHS + BLK - 1) / BLK, BLK, 0, stream>>>(
        gcnt, N_GRAPHS * D_HID + N_GRAPHS);
    k_pool<<<g_nf, BLK, 0, stream>>>(bufB, batch, gsum, gcnt);
    k_head<<<N_GRAPHS, 128, 0, stream>>>(gsum, gcnt, fc1_w, fc1_b, fc2_w, fc2_b, out);
}